// MILModel_90469191123190
// MI455X (gfx1250) — compile-verified
//
#include <hip/hip_runtime.h>
#include <hip/hip_bf16.h>
#include <math.h>

#define BB   16      // bags
#define NN   4096    // instances per bag
#define DD   1024    // feature dim
#define CC   4       // classes
#define HH   512     // hidden
#define X3D  3072    // 3*D
#define NCH  16      // chunks per bag
#define CHR  256     // rows per chunk (NN/NCH)

typedef __attribute__((ext_vector_type(2))) float v2f;
typedef __attribute__((ext_vector_type(8))) float v8f;

// ---------------------------------------------------------------------------
// Kernel 1: per-(bag,chunk) streaming pass.
//   Phase A: scores via wave-per-row dot(bags_row, query), wave32 xor-shuffle
//            reduction. Chunk (1 MB) stays L2-resident for phase B.
//   Phase B: column-parallel float4 stream computing chunk-local softmax
//            numerator v_ch, column sum (mean pool), column max (max pool).
// ---------------------------------------------------------------------------
__global__ __launch_bounds__(256) void mil_chunk_kernel(
    const float* __restrict__ bags, const float* __restrict__ query,
    float* __restrict__ chunkV, float* __restrict__ chunkSum,
    float* __restrict__ chunkMax, float* __restrict__ mArr,
    float* __restrict__ sArr) {
  __shared__ float qs[DD];
  __shared__ float sc[CHR];   // scores, then unnormalized weights e^(s-m)
  __shared__ float red[8];

  const int tid  = threadIdx.x;
  const int lane = tid & 31;
  const int wave = tid >> 5;
  const int b    = blockIdx.x / NCH;
  const int ch   = blockIdx.x % NCH;
  const float* base = bags + ((size_t)b * NN + (size_t)ch * CHR) * DD;

  for (int i = tid; i < DD; i += 256) qs[i] = query[i];
  __syncthreads();

  // ---- Phase A: dot-product scores (wave per row) ----
  for (int r = wave; r < CHR; r += 8) {
    const float* row = base + (size_t)r * DD;
    float acc = 0.f;
#pragma unroll
    for (int j = 0; j < DD / 32; ++j) {
      const int d = lane + 32 * j;
      acc = fmaf(row[d], qs[d], acc);
    }
#pragma unroll
    for (int off = 16; off > 0; off >>= 1) acc += __shfl_xor(acc, off, 32);
    if (lane == 0) sc[r] = acc;
  }
  __syncthreads();

  // ---- chunk-local max m and sum s of e^(score-m) ----
  const float v = sc[tid];
  float m = v;
#pragma unroll
  for (int off = 16; off > 0; off >>= 1) m = fmaxf(m, __shfl_xor(m, off, 32));
  if (lane == 0) red[wave] = m;
  __syncthreads();
  m = red[0];
#pragma unroll
  for (int w = 1; w < 8; ++w) m = fmaxf(m, red[w]);
  __syncthreads();  // everyone has read red[] for max before it is rewritten

  const float e = __expf(v - m);
  float s = e;
#pragma unroll
  for (int off = 16; off > 0; off >>= 1) s += __shfl_xor(s, off, 32);
  if (lane == 0) red[wave] = s;
  sc[tid] = e;  // each thread overwrites only its own slot
  __syncthreads();
  s = 0.f;
#pragma unroll
  for (int w = 0; w < 8; ++w) s += red[w];
  if (tid == 0) {
    mArr[b * NCH + ch] = m;
    sArr[b * NCH + ch] = s;
  }

  // ---- Phase B: column-parallel accumulation (4 consecutive cols/thread) ----
  float va0 = 0.f, va1 = 0.f, va2 = 0.f, va3 = 0.f;
  float su0 = 0.f, su1 = 0.f, su2 = 0.f, su3 = 0.f;
  float mx0 = -INFINITY, mx1 = -INFINITY, mx2 = -INFINITY, mx3 = -INFINITY;
  const float4* base4 = (const float4*)base;
  for (int r = 0; r < CHR; ++r) {
    const float w = sc[r];
    const float4 val = base4[(size_t)r * (DD / 4) + tid];
    // prefetch 16 rows ahead (speculative; OOB translation failures dropped)
    __builtin_prefetch(base + (size_t)(r + 16) * DD + 4 * tid, 0, 0);
    va0 = fmaf(w, val.x, va0); va1 = fmaf(w, val.y, va1);
    va2 = fmaf(w, val.z, va2); va3 = fmaf(w, val.w, va3);
    su0 += val.x; su1 += val.y; su2 += val.z; su3 += val.w;
    mx0 = fmaxf(mx0, val.x); mx1 = fmaxf(mx1, val.y);
    mx2 = fmaxf(mx2, val.z); mx3 = fmaxf(mx3, val.w);
  }
  const size_t o4 = ((size_t)(b * NCH + ch)) * (DD / 4) + tid;
  ((float4*)chunkV)[o4]   = make_float4(va0, va1, va2, va3);
  ((float4*)chunkSum)[o4] = make_float4(su0, su1, su2, su3);
  ((float4*)chunkMax)[o4] = make_float4(mx0, mx1, mx2, mx3);
}

// ---------------------------------------------------------------------------
// Kernel 2: per-bag combine of chunk partials -> x = [mean | max | attn]
// ---------------------------------------------------------------------------
__global__ __launch_bounds__(256) void mil_combine_kernel(
    const float* __restrict__ chunkV, const float* __restrict__ chunkSum,
    const float* __restrict__ chunkMax, const float* __restrict__ mArr,
    const float* __restrict__ sArr, float* __restrict__ x) {
  const int b = blockIdx.x;
  const int tid = threadIdx.x;

  float m = -INFINITY;
#pragma unroll
  for (int ch = 0; ch < NCH; ++ch) m = fmaxf(m, mArr[b * NCH + ch]);
  float scale[NCH];
  float Z = 0.f;
#pragma unroll
  for (int ch = 0; ch < NCH; ++ch) {
    scale[ch] = __expf(mArr[b * NCH + ch] - m);
    Z = fmaf(scale[ch], sArr[b * NCH + ch], Z);
  }
  const float invZ = 1.0f / Z;

  float a0 = 0.f, a1 = 0.f, a2 = 0.f, a3 = 0.f;
  float s0 = 0.f, s1 = 0.f, s2 = 0.f, s3 = 0.f;
  float x0 = -INFINITY, x1 = -INFINITY, x2 = -INFINITY, x3 = -INFINITY;
#pragma unroll 4
  for (int ch = 0; ch < NCH; ++ch) {
    const size_t o4 = ((size_t)(b * NCH + ch)) * (DD / 4) + tid;
    const float4 vv = ((const float4*)chunkV)[o4];
    const float4 ss = ((const float4*)chunkSum)[o4];
    const float4 mm = ((const float4*)chunkMax)[o4];
    const float sc = scale[ch];
    a0 = fmaf(sc, vv.x, a0); a1 = fmaf(sc, vv.y, a1);
    a2 = fmaf(sc, vv.z, a2); a3 = fmaf(sc, vv.w, a3);
    s0 += ss.x; s1 += ss.y; s2 += ss.z; s3 += ss.w;
    x0 = fmaxf(x0, mm.x); x1 = fmaxf(x1, mm.y);
    x2 = fmaxf(x2, mm.z); x3 = fmaxf(x3, mm.w);
  }
  const float invN = 1.0f / (float)NN;
  float* xb = x + (size_t)b * X3D;
  const int d = 4 * tid;
  ((float4*)(xb + d))[0]            = make_float4(s0 * invN, s1 * invN, s2 * invN, s3 * invN);
  ((float4*)(xb + DD + d))[0]       = make_float4(x0, x1, x2, x3);
  ((float4*)(xb + 2 * DD + d))[0]   = make_float4(a0 * invZ, a1 * invZ, a2 * invZ, a3 * invZ);
}

// ---------------------------------------------------------------------------
// Kernel 3: MLP head with fp32 WMMA (V_WMMA_F32_16X16X4_F32).
//   Batch M = 16 maps exactly onto one WMMA tile row.
//   GEMM1: [16,3072] @ [3072,512] + b1  -> LDS y (8 waves x 4 N-tiles)
//   GEMM2: [16,512]  @ [512,4]    + b2  -> out (wave 0, N padded to 16)
// ---------------------------------------------------------------------------
__global__ __launch_bounds__(256) void mil_mlp_wmma_kernel(
    const float* __restrict__ x, const float* __restrict__ W1,
    const float* __restrict__ b1, const float* __restrict__ W2,
    const float* __restrict__ b2, float* __restrict__ out) {
  __shared__ float y[16][HH];  // 32 KB

  const int tid  = threadIdx.x;
  const int lane = tid & 31;
  const int wave = tid >> 5;
  const int half = lane >> 4;   // 0: lanes 0-15, 1: lanes 16-31
  const int l16  = lane & 15;

  // ---- GEMM1 ----
  for (int tt = 0; tt < 4; ++tt) {
    const int n0 = wave * 64 + tt * 16;
    v8f c = {};
    for (int k0 = 0; k0 < X3D; k0 += 4) {
      const int ka = k0 + 2 * half;
      // A frag (16x4 fp32): lane L<16 -> M=L,K=k0..k0+1 ; L>=16 -> K=k0+2..k0+3
      v2f a;
      a.x = x[(size_t)l16 * X3D + ka];
      a.y = x[(size_t)l16 * X3D + ka + 1];
      // B frag (4x16 fp32): row-striped across lanes
      v2f bf;
      bf.x = W1[(size_t)ka * HH + n0 + l16];
      bf.y = W1[(size_t)(ka + 1) * HH + n0 + l16];
      c = __builtin_amdgcn_wmma_f32_16x16x4_f32(false, a, false, bf,
                                                (short)0, c, false, false);
    }
    const float bias = b1[n0 + l16];
#pragma unroll
    for (int r = 0; r < 8; ++r) y[8 * half + r][n0 + l16] = c[r] + bias;
  }
  __syncthreads();

  // ---- GEMM2 (wave 0; full EXEC within the wave as WMMA requires) ----
  if (wave == 0) {
    v8f c = {};
    for (int k0 = 0; k0 < HH; k0 += 4) {
      const int ka = k0 + 2 * half;
      v2f a;
      a.x = y[l16][ka];
      a.y = y[l16][ka + 1];
      v2f bf;
      bf.x = (l16 < CC) ? W2[ka * CC + l16] : 0.f;
      bf.y = (l16 < CC) ? W2[(ka + 1) * CC + l16] : 0.f;
      c = __builtin_amdgcn_wmma_f32_16x16x4_f32(false, a, false, bf,
                                                (short)0, c, false, false);
    }
    if (l16 < CC) {
      const float bias = b2[l16];
#pragma unroll
      for (int r = 0; r < 8; ++r) out[(8 * half + r) * CC + l16] = c[r] + bias;
    }
  }
}

// ---------------------------------------------------------------------------
extern "C" void kernel_launch(void* const* d_in, const int* in_sizes, int n_in,
                              void* d_out, int out_size, void* d_ws, size_t ws_size,
                              hipStream_t stream) {
  const float* bags  = (const float*)d_in[0];
  const float* query = (const float*)d_in[1];
  const float* W1    = (const float*)d_in[2];
  const float* b1    = (const float*)d_in[3];
  const float* W2    = (const float*)d_in[4];
  const float* b2    = (const float*)d_in[5];
  float* out = (float*)d_out;

  // workspace layout (floats)
  float* ws       = (float*)d_ws;
  float* chunkV   = ws;                               // 16*16*1024
  float* chunkSum = chunkV   + (size_t)BB * NCH * DD; // 16*16*1024
  float* chunkMax = chunkSum + (size_t)BB * NCH * DD; // 16*16*1024
  float* mArr     = chunkMax + (size_t)BB * NCH * DD; // 256
  float* sArr     = mArr + BB * NCH;                  // 256
  float* x        = sArr + BB * NCH;                  // 16*3072

  mil_chunk_kernel<<<BB * NCH, 256, 0, stream>>>(bags, query, chunkV, chunkSum,
                                                 chunkMax, mArr, sArr);
  mil_combine_kernel<<<BB, 256, 0, stream>>>(chunkV, chunkSum, chunkMax, mArr,
                                             sArr, x);
  mil_mlp_wmma_kernel<<<1, 256, 0, stream>>>(x, W1, b1, W2, b2, out);
}